// GATv2_10041633538648
// MI455X (gfx1250) — compile-verified
//
#include <hip/hip_runtime.h>
#include <hip/hip_bf16.h>

// ---------------------------------------------------------------------------
// GATv2 (2 layers, H=2, C=128) + 2-layer MLP head, fp32 throughout.
// GEMMs use V_WMMA_F32_16X16X4_F32 fed from LDS-staged B tiles (320KB WGP LDS,
// bank-conflict-free padded stride). Edge phase exploits MI455X's 192MB L2:
// xl/xr/out are 51.2MB each, so gathers and scatter-atomics are L2-resident.
// ---------------------------------------------------------------------------

typedef __attribute__((ext_vector_type(2))) float v2f;
typedef __attribute__((ext_vector_type(8))) float v8f;

#ifndef GAT_SLOPE
#define GAT_SLOPE 0.2f
#endif

#define BSTRIDE 80  // LDS row stride in floats (64 data + 16 pad -> 32 banks)

// Order-preserving float->uint encoding so atomicMax(uint) == float max.
__device__ __forceinline__ unsigned fenc(float f) {
  unsigned u = __float_as_uint(f);
  return (u & 0x80000000u) ? ~u : (u | 0x80000000u);
}
__device__ __forceinline__ float fdec(unsigned u) {
  return (u & 0x80000000u) ? __uint_as_float(u & 0x7FFFFFFFu)
                           : __uint_as_float(~u);
}

// ---------------------------------------------------------------------------
// fp32 WMMA GEMM: C[M,N] = A[M,K] @ B[K,N] (+bias[N]).
// Workgroup = 8 waves; all waves share one 64-column B strip staged in LDS.
// Each wave computes a 32x64 tile: 2 A-fragments x 4 B-fragments -> 8 WMMA
// accumulators, 8 v_wmma per k-step.
// A-frag (V_WMMA_F32_16X16X4_F32): lane<16 holds K=k,k+1 for M=lane;
// lane>=16 holds K=k+2,k+3 for M=lane-16. B rows striped across lanes.
// Requires N % 64 == 0, K % 4 == 0.
// ---------------------------------------------------------------------------
template <bool ADD_BIAS>
__global__ __launch_bounds__(256) void wmma_gemm_f32(
    const float* __restrict__ A, const float* __restrict__ B,
    const float* __restrict__ bias, float* __restrict__ C,
    int M, int K, int N) {
  extern __shared__ float Bs[];  // [K][BSTRIDE], 64 valid columns per row

  const int tid   = threadIdx.x;
  const int lane  = tid & 31;
  const int wave  = tid >> 5;
  const int tilesN = N >> 6;
  const int tileN  = blockIdx.x % tilesN;
  const int mGroup = blockIdx.x / tilesN;
  const int col0   = tileN << 6;

  // Cooperative stage of the K x 64 B strip into LDS (float4 granularity).
  const int nf4 = K << 4;  // K * 16 float4s
  float4* Bs4 = reinterpret_cast<float4*>(Bs);
  for (int i = tid; i < nf4; i += 256) {
    const int k  = i >> 4;
    const int c4 = i & 15;
    Bs4[k * (BSTRIDE / 4) + c4] =
        *reinterpret_cast<const float4*>(B + (size_t)k * N + col0 + c4 * 4);
  }
  __syncthreads();

  const int row0 = (mGroup * 8 + wave) * 32;
  if (row0 >= M) return;  // wave-uniform

  const int r  = lane & 15;
  const int hi = lane >> 4;  // 0 or 1
  int ra0 = row0 + r;       if (ra0 > M - 1) ra0 = M - 1;  // clamp tail loads
  int ra1 = row0 + 16 + r;  if (ra1 > M - 1) ra1 = M - 1;

  v8f acc[2][4];
#pragma unroll
  for (int p = 0; p < 2; ++p)
#pragma unroll
    for (int j = 0; j < 4; ++j) acc[p][j] = v8f{};

  const float* A0 = A + (size_t)ra0 * K + 2 * hi;
  const float* A1 = A + (size_t)ra1 * K + 2 * hi;
  const float* Bl = Bs + r;

  for (int k = 0; k < K; k += 4) {
    v2f a0, a1;
    { const float2 t = *reinterpret_cast<const float2*>(A0 + k); a0.x = t.x; a0.y = t.y; }
    { const float2 t = *reinterpret_cast<const float2*>(A1 + k); a1.x = t.x; a1.y = t.y; }
    const float* bp0 = Bl + (size_t)(k + hi) * BSTRIDE;  // rows k+hi
    const float* bp1 = bp0 + 2 * BSTRIDE;                // rows k+2+hi
#pragma unroll
    for (int j = 0; j < 4; ++j) {
      v2f b;
      b.x = bp0[j * 16];
      b.y = bp1[j * 16];
      acc[0][j] = __builtin_amdgcn_wmma_f32_16x16x4_f32(
          false, a0, false, b, (short)0, acc[0][j], false, false);
      acc[1][j] = __builtin_amdgcn_wmma_f32_16x16x4_f32(
          false, a1, false, b, (short)0, acc[1][j], false, false);
    }
  }

#pragma unroll
  for (int p = 0; p < 2; ++p) {
#pragma unroll
    for (int j = 0; j < 4; ++j) {
      const int cc = col0 + j * 16 + r;
      const float bv = ADD_BIAS ? bias[cc] : 0.0f;
#pragma unroll
      for (int i = 0; i < 8; ++i) {
        const int rr = row0 + p * 16 + i + 8 * hi;
        if (rr < M) C[(size_t)rr * N + cc] = acc[p][j][i] + bv;
      }
    }
  }
}

// ---------------------------------------------------------------------------
// Per-edge attention scores. One wave per edge; each lane covers 4 channels
// of head0 and 4 of head1. score[e,h] = att[h] . leaky_relu(xl[src]+xr[dst]).
// Segment-max into nmax[dst,h] via order-preserving atomicMax.
// ---------------------------------------------------------------------------
__global__ __launch_bounds__(256) void gat_edge_score(
    const float* __restrict__ xl, const float* __restrict__ xr,
    const float* __restrict__ att, const int* __restrict__ src,
    const int* __restrict__ dst, float* __restrict__ score,
    unsigned* __restrict__ nmax, int E) {
  const int lane = threadIdx.x & 31;
  const int e = (blockIdx.x * blockDim.x + threadIdx.x) >> 5;
  if (e >= E) return;
  const int s = src[e], d = dst[e];
  const float4* xls = reinterpret_cast<const float4*>(xl + (size_t)s * 256);
  const float4* xrd = reinterpret_cast<const float4*>(xr + (size_t)d * 256);
  const float4* at  = reinterpret_cast<const float4*>(att);

  float p0, p1;
  {
    float4 a = xls[lane], b = xrd[lane], w = at[lane];
    float vx = a.x + b.x, vy = a.y + b.y, vz = a.z + b.z, vw = a.w + b.w;
    vx = vx > 0.f ? vx : GAT_SLOPE * vx;
    vy = vy > 0.f ? vy : GAT_SLOPE * vy;
    vz = vz > 0.f ? vz : GAT_SLOPE * vz;
    vw = vw > 0.f ? vw : GAT_SLOPE * vw;
    p0 = w.x * vx + w.y * vy + w.z * vz + w.w * vw;
  }
  {
    float4 a = xls[lane + 32], b = xrd[lane + 32], w = at[lane + 32];
    float vx = a.x + b.x, vy = a.y + b.y, vz = a.z + b.z, vw = a.w + b.w;
    vx = vx > 0.f ? vx : GAT_SLOPE * vx;
    vy = vy > 0.f ? vy : GAT_SLOPE * vy;
    vz = vz > 0.f ? vz : GAT_SLOPE * vz;
    vw = vw > 0.f ? vw : GAT_SLOPE * vw;
    p1 = w.x * vx + w.y * vy + w.z * vz + w.w * vw;
  }
#pragma unroll
  for (int off = 16; off > 0; off >>= 1) {
    p0 += __shfl_xor(p0, off, 32);
    p1 += __shfl_xor(p1, off, 32);
  }
  if (lane == 0) {
    score[2 * e]     = p0;
    score[2 * e + 1] = p1;
    atomicMax(&nmax[2 * d],     fenc(p0));
    atomicMax(&nmax[2 * d + 1], fenc(p1));
  }
}

// a = exp(score - max[dst]); denom[dst] += a; score overwritten with a.
__global__ __launch_bounds__(256) void gat_edge_exp(
    const int* __restrict__ dst, const unsigned* __restrict__ nmax,
    float* __restrict__ score, float* __restrict__ denom, int E) {
  const int i = blockIdx.x * blockDim.x + threadIdx.x;
  if (i >= 2 * E) return;
  const int e = i >> 1, h = i & 1;
  const int d = dst[e];
  const float m = fdec(nmax[2 * d + h]);
  const float a = __expf(score[i] - m);
  score[i] = a;
  atomicAdd(&denom[2 * d + h], a);
}

// out[dst] += (a / (denom+eps)) * xl[src]; one wave per edge, float atomics
// into the L2-resident accumulator.
__global__ __launch_bounds__(256) void gat_edge_aggregate(
    const float* __restrict__ xl, const float* __restrict__ score,
    const float* __restrict__ denom, const int* __restrict__ src,
    const int* __restrict__ dst, float* __restrict__ out, int E) {
  const int lane = threadIdx.x & 31;
  const int e = (blockIdx.x * blockDim.x + threadIdx.x) >> 5;
  if (e >= E) return;
  const int s = src[e], d = dst[e];
  const float a0 = score[2 * e]     / (denom[2 * d]     + 1e-16f);
  const float a1 = score[2 * e + 1] / (denom[2 * d + 1] + 1e-16f);
  const float4* xs = reinterpret_cast<const float4*>(xl + (size_t)s * 256);
  const float4 v0 = xs[lane];
  const float4 v1 = xs[lane + 32];
  float* o0 = out + (size_t)d * 256 + lane * 4;
  float* o1 = o0 + 128;
  atomicAdd(o0 + 0, a0 * v0.x);
  atomicAdd(o0 + 1, a0 * v0.y);
  atomicAdd(o0 + 2, a0 * v0.z);
  atomicAdd(o0 + 3, a0 * v0.w);
  atomicAdd(o1 + 0, a1 * v1.x);
  atomicAdd(o1 + 1, a1 * v1.y);
  atomicAdd(o1 + 2, a1 * v1.z);
  atomicAdd(o1 + 3, a1 * v1.w);
}

// h[i] = act(h[i] + b[i & cmask])
template <bool RELU>
__global__ __launch_bounds__(256) void bias_act(
    float* __restrict__ h, const float* __restrict__ b, int total, int cmask) {
  const int i = blockIdx.x * blockDim.x + threadIdx.x;
  if (i >= total) return;
  float v = h[i] + b[i & cmask];
  if (RELU) v = v > 0.f ? v : 0.f;
  h[i] = v;
}

// out[n,o] = sigmoid(h3[n,:] . W4[:,o] + b4[o]), OUT=40, K=128.
__global__ __launch_bounds__(256) void final_mlp(
    const float* __restrict__ h3, const float* __restrict__ W4,
    const float* __restrict__ b4, float* __restrict__ out, int Nn) {
  const int i = blockIdx.x * blockDim.x + threadIdx.x;
  if (i >= Nn * 40) return;
  const int n = i / 40, o = i - n * 40;
  const float* row = h3 + (size_t)n * 128;
  float acc = b4[o];
#pragma unroll 8
  for (int k = 0; k < 128; ++k) acc += row[k] * W4[k * 40 + o];
  out[i] = 1.0f / (1.0f + __expf(-acc));
}

// ---------------------------------------------------------------------------

static inline void launch_gemm(const float* A, const float* B, const float* bias,
                               float* C, int M, int K, int N, bool addBias,
                               hipStream_t stream) {
  const int tilesN  = N / 64;
  const int mGroups = (M + 255) / 256;  // 8 waves x 32 rows per block
  const int blocks  = mGroups * tilesN;
  const size_t shmem = (size_t)K * BSTRIDE * sizeof(float);
  if (addBias)
    wmma_gemm_f32<true><<<blocks, 256, shmem, stream>>>(A, B, bias, C, M, K, N);
  else
    wmma_gemm_f32<false><<<blocks, 256, shmem, stream>>>(A, B, nullptr, C, M, K, N);
}

extern "C" void kernel_launch(void* const* d_in, const int* in_sizes, int n_in,
                              void* d_out, int out_size, void* d_ws,
                              size_t ws_size, hipStream_t stream) {
  (void)n_in; (void)out_size; (void)ws_size;

  const float* x   = (const float*)d_in[0];
  const int*   ei  = (const int*)d_in[1];   // int32 in practice (JAX x64 off)
  const float* Wl1 = (const float*)d_in[2];
  const float* Wr1 = (const float*)d_in[3];
  const float* at1 = (const float*)d_in[4];
  const float* b1  = (const float*)d_in[5];
  const float* Wl2 = (const float*)d_in[6];
  const float* Wr2 = (const float*)d_in[7];
  const float* at2 = (const float*)d_in[8];
  const float* b2  = (const float*)d_in[9];
  const float* W3  = (const float*)d_in[10];
  const float* b3  = (const float*)d_in[11];
  const float* W4  = (const float*)d_in[12];
  const float* b4  = (const float*)d_in[13];

  const int E  = in_sizes[1] / 2;    // 650000
  const int Nn = in_sizes[0] / 128;  // 50000
  const int* src = ei;
  const int* dst = ei + E;

  // Workspace layout (bytes): XL | XR | H | SCORE | NMAX | DENOM   (~160 MB)
  char* ws = (char*)d_ws;
  const size_t szF = (size_t)Nn * 256 * sizeof(float);  // 51.2 MB
  float*    XL  = (float*)ws;
  float*    XR  = (float*)(ws + szF);
  float*    H   = (float*)(ws + 2 * szF);
  float*    SC  = (float*)(ws + 3 * szF);
  unsigned* NM  = (unsigned*)(ws + 3 * szF + (size_t)E * 2 * sizeof(float));
  float*    DEN = (float*)((char*)NM + (size_t)Nn * 2 * sizeof(unsigned));

  const int edgeBlocks  = (E + 7) / 8;               // 1 wave/edge, 8 waves/blk
  const int edge2Blocks = (2 * E + 255) / 256;
  const int featBlocks  = (Nn * 256 + 255) / 256;

  // ---------------- Layer 1 ----------------
  launch_gemm(x, Wl1, nullptr, XL, Nn, 128, 256, false, stream);
  launch_gemm(x, Wr1, nullptr, XR, Nn, 128, 256, false, stream);
  hipMemsetAsync(NM, 0, (size_t)Nn * 2 * sizeof(unsigned), stream);
  hipMemsetAsync(DEN, 0, (size_t)Nn * 2 * sizeof(float), stream);
  hipMemsetAsync(H, 0, szF, stream);
  gat_edge_score<<<edgeBlocks, 256, 0, stream>>>(XL, XR, at1, src, dst, SC, NM, E);
  gat_edge_exp<<<edge2Blocks, 256, 0, stream>>>(dst, NM, SC, DEN, E);
  gat_edge_aggregate<<<edgeBlocks, 256, 0, stream>>>(XL, SC, DEN, src, dst, H, E);
  bias_act<true><<<featBlocks, 256, 0, stream>>>(H, b1, Nn * 256, 255);

  // ---------------- Layer 2 ----------------
  launch_gemm(H, Wl2, nullptr, XL, Nn, 256, 256, false, stream);
  launch_gemm(H, Wr2, nullptr, XR, Nn, 256, 256, false, stream);
  hipMemsetAsync(NM, 0, (size_t)Nn * 2 * sizeof(unsigned), stream);
  hipMemsetAsync(DEN, 0, (size_t)Nn * 2 * sizeof(float), stream);
  hipMemsetAsync(H, 0, szF, stream);
  gat_edge_score<<<edgeBlocks, 256, 0, stream>>>(XL, XR, at2, src, dst, SC, NM, E);
  gat_edge_exp<<<edge2Blocks, 256, 0, stream>>>(dst, NM, SC, DEN, E);
  gat_edge_aggregate<<<edgeBlocks, 256, 0, stream>>>(XL, SC, DEN, src, dst, H, E);
  bias_act<true><<<featBlocks, 256, 0, stream>>>(H, b2, Nn * 256, 255);

  // ---------------- Head: W3 (+b3) then W4 + sigmoid ----------------
  launch_gemm(H, W3, b3, XL, Nn, 256, 128, true, stream);  // h3 -> XL (25.6MB)
  const int outBlocks = (Nn * 40 + 255) / 256;
  final_mlp<<<outBlocks, 256, 0, stream>>>(XL, W4, b4, (float*)d_out, Nn);
}